// UnsupervisedGCN_11398843204159
// MI455X (gfx1250) — compile-verified
//
#include <hip/hip_runtime.h>

#define HID      64
#define NNODES   100000
#define NEDGES   1200000
#define NGRAPHS  1000

typedef __attribute__((ext_vector_type(2))) float v2f;
typedef __attribute__((ext_vector_type(8))) float v8f;

// ---------------------------------------------------------------------------
// GEMM: Out[N,64] = act(In[N,64]) @ W[64,64]
//   act = (+bias, relu) applied to the INPUT when RELU_IN (fuses layer-1
//   epilogue into layer-2's A-tile load).
// Block = 128 threads = 4 waves; block covers 16 rows, wave w covers cols
// [16w, 16w+16) via V_WMMA_F32_16X16X4_F32 accumulating over K=64 in steps
// of 4.  f32 WMMA keeps full reference precision; GEMM cost is negligible.
// ---------------------------------------------------------------------------
template <bool RELU_IN>
__global__ __launch_bounds__(128) void gcn_gemm_wmma(
    const float* __restrict__ In, const float* __restrict__ W,
    const float* __restrict__ bias_in, float* __restrict__ Out)
{
    __shared__ float sW[64 * 65];  // stride-65 padding: conflict-free reads
    __shared__ float sA[16 * 65];

    const int tid  = threadIdx.x;
    const int wave = tid >> 5;
    const int lane = tid & 31;
    const int rowBase = blockIdx.x * 16;   // 100000 % 16 == 0, no tail

    // stage W (64x64) into LDS (coalesced global reads)
    for (int i = tid; i < 64 * 64; i += 128) {
        int r = i >> 6, c = i & 63;
        sW[r * 65 + c] = W[i];
    }
    // stage A tile (16x64), fusing bias+relu of previous layer if requested
    for (int i = tid; i < 16 * 64; i += 128) {
        int r = i >> 6, c = i & 63;
        float v = In[(rowBase + r) * 64 + c];
        if (RELU_IN) {
            v += bias_in[c];
            v = v > 0.0f ? v : 0.0f;
        }
        sA[r * 65 + c] = v;
    }
    __syncthreads();

    // WMMA f32 16x16x4 operand placement (wave32):
    //  A: lane L holds row m=L%16; VGPR0=A[m][k + 2*(L/16)], VGPR1=A[m][k+1+2*(L/16)]
    //  B: lane L holds col n=L%16 of rows k+2*(L/16), k+1+2*(L/16)
    //  C/D: VGPR v holds element (m = v + 8*(L/16), n = L%16)
    const int m     = lane & 15;
    const int n     = (wave << 4) + (lane & 15);
    const int khalf = (lane >> 4) << 1;  // 0 or 2

    v8f acc = {};
#pragma unroll
    for (int k = 0; k < 64; k += 4) {
        v2f a, b;
        a.x = sA[m * 65 + k + khalf];
        a.y = sA[m * 65 + k + khalf + 1];
        b.x = sW[(k + khalf) * 65 + n];
        b.y = sW[(k + khalf + 1) * 65 + n];
        acc = __builtin_amdgcn_wmma_f32_16x16x4_f32(
            /*neg_a=*/false, a, /*neg_b=*/false, b,
            /*c_mod=*/(short)0, acc, /*reuse_a=*/false, /*reuse_b=*/false);
    }

    const int rOff = (lane >> 4) << 3;  // 0 or 8
    float* outp = Out + (size_t)(rowBase + rOff) * 64 + n;
#pragma unroll
    for (int v = 0; v < 8; ++v)
        outp[(size_t)v * 64] = acc[v];
}

// ---------------------------------------------------------------------------
// Edge scatter (SpMM): Agg[dst] += X[src].  One wave32 per edge: src/dst are
// wave-uniform (scalar loads), each lane moves a float2 (coalesced 256B
// gather) and issues two hardware global_atomic_add_f32 (L2-resident).
// ---------------------------------------------------------------------------
__global__ __launch_bounds__(256) void gcn_scatter(
    const float* __restrict__ X, const int* __restrict__ src,
    const int* __restrict__ dst, float* __restrict__ Agg)
{
    const unsigned gid = blockIdx.x * 256u + threadIdx.x;
    const unsigned e = gid >> 5;          // edge index (wave-uniform)
    const unsigned l = gid & 31;          // lane -> column pair
    if (e >= NEDGES) return;
    const int s = src[e];
    const int d = dst[e];
    const float2 v = *(const float2*)(X + (size_t)s * 64 + l * 2);
    float* p = Agg + (size_t)d * 64 + l * 2;
    unsafeAtomicAdd(p, v.x);
    unsafeAtomicAdd(p + 1, v.y);
}

// ---------------------------------------------------------------------------
// Readout accumulation: per (node, col) atomic add into graph sums; per node
// count.  1000*64 distinct sum addresses -> all resolved in L2 atomic units.
// ---------------------------------------------------------------------------
__global__ __launch_bounds__(256) void gcn_readout_acc(
    const float* __restrict__ Agg2, const int* __restrict__ gids,
    float* __restrict__ gsum, float* __restrict__ gcnt)
{
    const unsigned gid = blockIdx.x * 256u + threadIdx.x;
    const unsigned node = gid >> 6;
    if (node >= NNODES) return;
    const unsigned col = gid & 63;
    const int g = gids[node];
    unsafeAtomicAdd(&gsum[(size_t)g * 64 + col], Agg2[gid]);
    if (col == 0) unsafeAtomicAdd(&gcnt[g], 1.0f);
}

// out = gsum/cnt + b2  (cnt>0), else 0 — algebraically identical to the
// reference's sum(agg2 + b2)/max(cnt,1).
__global__ __launch_bounds__(256) void gcn_finalize(
    const float* __restrict__ gsum, const float* __restrict__ gcnt,
    const float* __restrict__ b2, float* __restrict__ out)
{
    const unsigned gid = blockIdx.x * 256u + threadIdx.x;
    if (gid >= NGRAPHS * 64u) return;
    const unsigned g = gid >> 6;
    const unsigned c = gid & 63;
    const float cnt = gcnt[g];
    out[gid] = (cnt > 0.0f) ? (gsum[gid] / cnt + b2[c]) : 0.0f;
}

// ---------------------------------------------------------------------------
extern "C" void kernel_launch(void* const* d_in, const int* in_sizes, int n_in,
                              void* d_out, int out_size, void* d_ws, size_t ws_size,
                              hipStream_t stream)
{
    (void)in_sizes; (void)n_in; (void)out_size; (void)ws_size;

    const float* feats = (const float*)d_in[0];
    const float* W1    = (const float*)d_in[1];
    const float* b1    = (const float*)d_in[2];
    const float* W2    = (const float*)d_in[3];
    const float* b2    = (const float*)d_in[4];
    const int*   src   = (const int*)d_in[5];
    const int*   dst   = (const int*)d_in[6];
    const int*   gids  = (const int*)d_in[7];
    float* out = (float*)d_out;

    // workspace: two ping-pong node buffers + graph accumulators
    const size_t nodeElems = (size_t)NNODES * HID;           // 6.4M floats
    float* bufA = (float*)d_ws;                              // X1 / X2
    float* bufB = bufA + nodeElems;                          // agg1 / agg2
    float* gsum = bufB + nodeElems;                          // 64000 floats
    float* gcnt = gsum + (size_t)NGRAPHS * HID;              // 1000 floats

    const int gemmBlocks    = NNODES / 16;                       // 6250
    const int scatterBlocks = (NEDGES * 32 + 255) / 256;         // 150000
    const int readBlocks    = (NNODES * 64 + 255) / 256;         // 25000
    const int finBlocks     = (NGRAPHS * 64 + 255) / 256;        // 250

    // Layer 1: X1 = feats @ W1
    hipLaunchKernelGGL((gcn_gemm_wmma<false>), dim3(gemmBlocks), dim3(128), 0, stream,
                       feats, W1, (const float*)nullptr, bufA);
    // agg1 = scatter-add X1[src] -> dst
    hipMemsetAsync(bufB, 0, nodeElems * sizeof(float), stream);
    hipLaunchKernelGGL(gcn_scatter, dim3(scatterBlocks), dim3(256), 0, stream,
                       bufA, src, dst, bufB);
    // Layer 2: X2 = relu(agg1 + b1) @ W2   (epilogue fused into A-tile load)
    hipLaunchKernelGGL((gcn_gemm_wmma<true>), dim3(gemmBlocks), dim3(128), 0, stream,
                       bufB, W2, b1, bufA);
    // agg2 = scatter-add X2[src] -> dst   (bufB reusable after gemm2 read it)
    hipMemsetAsync(bufB, 0, nodeElems * sizeof(float), stream);
    hipLaunchKernelGGL(gcn_scatter, dim3(scatterBlocks), dim3(256), 0, stream,
                       bufA, src, dst, bufB);
    // Mean-pool readout (b2 folded into finalize)
    hipMemsetAsync(gsum, 0, ((size_t)NGRAPHS * HID + NGRAPHS) * sizeof(float), stream);
    hipLaunchKernelGGL(gcn_readout_acc, dim3(readBlocks), dim3(256), 0, stream,
                       bufB, gids, gsum, gcnt);
    hipLaunchKernelGGL(gcn_finalize, dim3(finBlocks), dim3(256), 0, stream,
                       gsum, gcnt, b2, out);
}